// TemporalMultiHeadAttention_28063316312871
// MI455X (gfx1250) — compile-verified
//
#include <hip/hip_runtime.h>
#include <math.h>

// ---------------------------------------------------------------------------
// TemporalMultiHeadAttention for MI455X (gfx1250, wave32)
//   - all matrix math on V_WMMA_F32_16X16X4_F32 (fp32 -> matches fp32 ref)
//   - attention K/V tiles staged LDS via Tensor Data Mover (double buffered)
// D=1024, H=16, DK=64, B=2, L=2048.
// ---------------------------------------------------------------------------

typedef float v2f __attribute__((ext_vector_type(2)));
typedef float v8f __attribute__((ext_vector_type(8)));
typedef unsigned int u32x4 __attribute__((ext_vector_type(4)));
typedef int i32x8 __attribute__((ext_vector_type(8)));
typedef int i32x4 __attribute__((ext_vector_type(4)));

#define D_MODEL   1024
#define NHEAD     16
#define DKH       64
#define SEQ       2048
#define BATCH     2
#define LAM_DECAY 0.1f
#define NEG_INF   -1e30f

#if defined(__gfx1250__) && __has_builtin(__builtin_amdgcn_tensor_load_to_lds)
#define USE_TDM 1
#else
#define USE_TDM 0
#endif

// D = A(16x4) * B(4x16) + C  (fp32, wave32)
__device__ __forceinline__ v8f wmma4(v2f a, v2f b, v8f c) {
  return __builtin_amdgcn_wmma_f32_16x16x4_f32(false, a, false, b, (short)0, c,
                                               false, false);
}

// Fragment layouts (ISA 7.12.2, wave32):
//  A 16x4 : lane l, vgpr v ->  m = l%16,          k = 2*(l/16) + v
//  B 4x16 : lane l, vgpr v ->  k = v + 2*(l/16),  n = l%16
//  C 16x16: lane l, vgpr v ->  m = v + 8*(l/16),  n = l%16

__device__ __forceinline__ unsigned lds_off(const void* p) {
  // generic pointer to LDS: low 32 bits are the LDS byte address
  return (unsigned)(unsigned long long)(uintptr_t)p;
}

#if USE_TDM
// TDM: copy 1024 contiguous f32 (one 16x64 K or V tile) from global to LDS,
// inserting 4 DWORDs of padding after every 64 DWORDs so the LDS image has
// row stride 68 floats (our bank-conflict-free layout).
__device__ __forceinline__ void tdm_load_tile(unsigned lds_addr,
                                              const void* gsrc) {
  const unsigned long long ga = (unsigned long long)(uintptr_t)gsrc;
  u32x4 g0;
  g0[0] = 1u;                                   // count=1 (valid descriptor)
  g0[1] = lds_addr;                             // lds_addr [63:32]
  g0[2] = (unsigned)ga;                         // global_addr[31:0]
  g0[3] = (unsigned)((ga >> 32) & 0x01ffffffu)  // global_addr[56:32]
          | (2u << 30);                         // type = 2 ("image")
  i32x8 g1;
  g1[0] = (int)((2u << 16)      // data_size = 4 bytes
              | (1u << 20)      // pad_enable
              | (5u << 22)      // pad_interval: 64 DWORDs
              | (3u << 25));    // pad_amount:   4 DWORDs
  g1[1] = (int)(1024u << 16);   // tensor_dim0 = 1024 (bits 79:48)
  g1[2] = (int)(1u << 16);      // tensor_dim0 hi | tensor_dim1 = 1
  g1[3] = (int)(1024u << 16);   // tensor_dim1 hi | tile_dim0 = 1024
  g1[4] = 1;                    // tile_dim1 = 1, tile_dim2 = 0
  g1[5] = 1024;                 // tensor_dim0_stride low
  g1[6] = 0;                    // stride0 hi | tensor_dim1_stride lo
  g1[7] = 0;
  const i32x4 gz4 = {0, 0, 0, 0};          // 2D tensor: groups 2/3 unused
  const i32x8 gz8 = {0, 0, 0, 0, 0, 0, 0, 0};
  // 6-arg form (this toolchain): (g0, g1, g2, g3, g4, cpol)
  __builtin_amdgcn_tensor_load_to_lds(g0, g1, gz4, gz4, gz8, 0);
}
#endif

// ---------------------------------------------------------------------------
// Kernel 1: fused QKV projection.  C[m,n] = sum_k X[m,k] * W[n,k]   (NT gemm)
// blockIdx.z selects Q/K/V.  Output layout [B,H,L,DK]; Q scaled by 1/sqrt(DK).
// ---------------------------------------------------------------------------
__global__ __launch_bounds__(256)
void qkv_proj_kernel(const float* __restrict__ X,
                     const float* __restrict__ Wq,
                     const float* __restrict__ Wk,
                     const float* __restrict__ Wv,
                     float* __restrict__ Qout,
                     float* __restrict__ Kout,
                     float* __restrict__ Vout) {
  const int z = blockIdx.z;
  const float* W   = (z == 0) ? Wq : (z == 1) ? Wk : Wv;
  float*       Out = (z == 0) ? Qout : (z == 1) ? Kout : Vout;
  const float  scale = (z == 0) ? 0.125f : 1.0f;  // 1/sqrt(64) folded into Q

  __shared__ float Asub[64][20];  // stride 20: float4-aligned, conflict-free
  __shared__ float Bsub[64][20];

  const int tid   = threadIdx.x;
  const int wave  = tid >> 5;
  const int lane  = tid & 31;
  const int lhalf = lane >> 4;
  const int l16   = lane & 15;

  const int mBase = blockIdx.y * 64;
  const int nBase = blockIdx.x * 64;
  const int mOff  = (wave >> 1) * 16;  // 4 waves along M
  const int nOff  = (wave & 1) * 32;   // 2 waves along N, 2 tiles each

  v8f acc0 = {};
  v8f acc1 = {};

  const int ldRow = tid >> 2;        // 0..63
  const int ldCol = (tid & 3) * 4;   // 0,4,8,12

  for (int k0 = 0; k0 < D_MODEL; k0 += 16) {
    const float4 av =
        *(const float4*)(X + (size_t)(mBase + ldRow) * D_MODEL + k0 + ldCol);
    const float4 bv =
        *(const float4*)(W + (size_t)(nBase + ldRow) * D_MODEL + k0 + ldCol);
    *(float4*)(&Asub[ldRow][ldCol]) = av;
    *(float4*)(&Bsub[ldRow][ldCol]) = bv;
    __syncthreads();

    if (k0 + 16 < D_MODEL) {  // -> global_prefetch_b8
      __builtin_prefetch(X + (size_t)(mBase + ldRow) * D_MODEL + k0 + 16 + ldCol, 0, 3);
      __builtin_prefetch(W + (size_t)(nBase + ldRow) * D_MODEL + k0 + 16 + ldCol, 0, 3);
    }

#pragma unroll
    for (int kk = 0; kk < 16; kk += 4) {
      const int kf = kk + 2 * lhalf;  // even -> 8B-aligned LDS float2
      v2f a  = *(const v2f*)(&Asub[mOff + l16][kf]);
      v2f b0 = *(const v2f*)(&Bsub[nOff + l16][kf]);
      v2f b1 = *(const v2f*)(&Bsub[nOff + 16 + l16][kf]);
      acc0 = wmma4(a, b0, acc0);
      acc1 = wmma4(a, b1, acc1);
    }
    __syncthreads();
  }

#pragma unroll
  for (int v = 0; v < 8; ++v) {
    const int m   = mBase + mOff + v + 8 * lhalf;  // token row in [0, B*L)
    const int bb  = m / SEQ;
    const int tok = m % SEQ;
    {
      const int col = nBase + nOff + l16;
      const int h = col >> 6, d = col & 63;
      Out[(((size_t)bb * NHEAD + h) * SEQ + tok) * DKH + d] = acc0[v] * scale;
    }
    {
      const int col = nBase + nOff + 16 + l16;
      const int h = col >> 6, d = col & 63;
      Out[(((size_t)bb * NHEAD + h) * SEQ + tok) * DKH + d] = acc1[v] * scale;
    }
  }
}

// ---------------------------------------------------------------------------
// Kernel 2: flash-attention with temporal/decay penalties + causal + pad mask.
// Block = 128 threads (4 waves); block handles 64 queries of one (b,h).
// K/V tiles staged in LDS by the Tensor Data Mover, double buffered so the
// DMA for tile i+1 overlaps the 32 WMMAs of tile i.
// ---------------------------------------------------------------------------
__global__ __launch_bounds__(128)
void attn_kernel(const float* __restrict__ Q, const float* __restrict__ K,
                 const float* __restrict__ V, const float* __restrict__ ts,
                 const float* __restrict__ decay,
                 const unsigned char* __restrict__ pmask,
                 const float* __restrict__ lambda_time_p,
                 float* __restrict__ Ctx) {
  __shared__ float sK[2][16][68];   // stride 68, matches TDM pad descriptor
  __shared__ float sV[2][16][68];
  __shared__ float sP[4][16][18];   // per-wave P tile (C-layout -> A-layout)

  const int b = blockIdx.z, h = blockIdx.y;
  const int qBase = blockIdx.x * 64;
  const int tid = threadIdx.x;
  const int wave = tid >> 5, lane = tid & 31;
  const int lhalf = lane >> 4, l16 = lane & 15;
  const float lt = lambda_time_p[0];

  const size_t headBase = ((size_t)b * NHEAD + h) * SEQ;

  // Q fragments for this wave's 16 rows (A-layout, DK=64 -> 16 k-steps)
  const int qRow = qBase + wave * 16 + l16;
  const float* qptr = Q + (headBase + qRow) * DKH;
  v2f qf[16];
#pragma unroll
  for (int kk = 0; kk < 16; ++kk)
    qf[kk] = *(const v2f*)(qptr + kk * 4 + 2 * lhalf);

  // per-row query timestamps (rows v + 8*lhalf in this lane's view)
  float tq[8];
#pragma unroll
  for (int v = 0; v < 8; ++v)
    tq[v] = ts[(size_t)b * SEQ + qBase + wave * 16 + v + 8 * lhalf];

  float mrow[8], lrow[8];
#pragma unroll
  for (int v = 0; v < 8; ++v) { mrow[v] = NEG_INF; lrow[v] = 0.f; }
  v8f o0 = {}, o1 = {}, o2 = {}, o3 = {};

  const int kEnd = qBase + 64;  // causal: keys <= last query of block

#if USE_TDM
  if (wave == 0) {  // prologue: DMA tile 0 into buffer 0 (TDM ignores EXEC)
    tdm_load_tile(lds_off(&sK[0][0][0]), K + headBase * DKH);
    tdm_load_tile(lds_off(&sV[0][0][0]), V + headBase * DKH);
  }
#endif

  int buf = 0;
  for (int kb = 0; kb < kEnd; kb += 16, buf ^= 1) {
#if USE_TDM
    if (wave == 0) {
      if (kb + 16 < kEnd) {
        // issue DMA for next tile into the other buffer, then wait for the
        // two oldest (current tile) transfers: TDM completes in order.
        tdm_load_tile(lds_off(&sK[buf ^ 1][0][0]),
                      K + (headBase + kb + 16) * DKH);
        tdm_load_tile(lds_off(&sV[buf ^ 1][0][0]),
                      V + (headBase + kb + 16) * DKH);
        __builtin_amdgcn_s_wait_tensorcnt(2);
      } else {
        __builtin_amdgcn_s_wait_tensorcnt(0);
      }
    }
    __syncthreads();
#else
    // fallback: cooperative synchronous load of the 16x64 K/V tiles
#pragma unroll
    for (int i = 0; i < 2; ++i) {
      const int idx = tid + i * 128;            // 0..255
      const int r = idx >> 4, c4 = (idx & 15) * 4;
      *(float4*)(&sK[buf][r][c4]) =
          *(const float4*)(K + (headBase + kb + r) * DKH + c4);
      *(float4*)(&sV[buf][r][c4]) =
          *(const float4*)(V + (headBase + kb + r) * DKH + c4);
    }
    __syncthreads();
#endif

    // per-key (column) penalties for this lane
    const int kj = kb + l16;
    const float tK   = ts[(size_t)b * SEQ + kj];
    const float dpen = LAM_DECAY * decay[(size_t)b * SEQ + kj];
    const bool keyok = pmask[(size_t)b * SEQ + kj] != 0;

    // S = (Q/sqrt(DK)) * K^T  : 16 fp32 WMMAs over DK=64
    v8f s = {};
#pragma unroll
    for (int kk = 0; kk < 16; ++kk) {
      v2f bf = *(const v2f*)(&sK[buf][l16][kk * 4 + 2 * lhalf]);
      s = wmma4(qf[kk], bf, s);
    }

    // online softmax per row (row lives in a 16-lane half -> shfl_xor tree)
    float fac[8];
#pragma unroll
    for (int v = 0; v < 8; ++v) {
      const int qi = qBase + wave * 16 + v + 8 * lhalf;
      float val = s[v] - lt * fabsf(tq[v] - tK) - dpen;
      if (kj > qi || !keyok) val = NEG_INF;

      float rmax = val;
#pragma unroll
      for (int off = 1; off < 16; off <<= 1)
        rmax = fmaxf(rmax, __shfl_xor(rmax, off, 32));
      const float newm = fmaxf(mrow[v], rmax);
      const float p = (val <= -1e29f) ? 0.f : __expf(val - newm);
      const float f = (mrow[v] <= -1e29f) ? 0.f : __expf(mrow[v] - newm);
      float rsum = p;
#pragma unroll
      for (int off = 1; off < 16; off <<= 1)
        rsum += __shfl_xor(rsum, off, 32);
      lrow[v] = lrow[v] * f + rsum;
      mrow[v] = newm;
      fac[v] = f;
      sP[wave][v + 8 * lhalf][l16] = p;  // C-layout -> LDS
    }

    // rescale O accumulators (same row<->vgpr mapping as S)
#pragma unroll
    for (int v = 0; v < 8; ++v) {
      o0[v] *= fac[v]; o1[v] *= fac[v]; o2[v] *= fac[v]; o3[v] *= fac[v];
    }

    // O += P * V : P re-read in A-layout; 4 k-steps x 4 d-tiles
#pragma unroll
    for (int kk = 0; kk < 4; ++kk) {
      v2f a = *(const v2f*)(&sP[wave][l16][kk * 4 + 2 * lhalf]);
#pragma unroll
      for (int t = 0; t < 4; ++t) {
        v2f bf;
        bf[0] = sV[buf][kk * 4 + 2 * lhalf + 0][t * 16 + l16];
        bf[1] = sV[buf][kk * 4 + 2 * lhalf + 1][t * 16 + l16];
        v8f& oo = (t == 0) ? o0 : (t == 1) ? o1 : (t == 2) ? o2 : o3;
        oo = wmma4(a, bf, oo);
      }
    }
    __syncthreads();  // all reads of sK/sV[buf] done before TDM reuses it
  }

  // epilogue: ctx[b, qi, h*64 + d] = O / l
#pragma unroll
  for (int v = 0; v < 8; ++v) {
    const int qi = qBase + wave * 16 + v + 8 * lhalf;
    const float inv = (lrow[v] > 0.f) ? (1.f / lrow[v]) : 0.f;
    float* dst = Ctx + ((size_t)b * SEQ + qi) * D_MODEL + h * DKH;
    dst[0 * 16 + l16] = o0[v] * inv;
    dst[1 * 16 + l16] = o1[v] * inv;
    dst[2 * 16 + l16] = o2[v] * inv;
    dst[3 * 16 + l16] = o3[v] * inv;
  }
}

// ---------------------------------------------------------------------------
// Kernel 3: output projection  out[m,n] = sum_k ctx[m,k]*Wo[n,k] + b[n]
// ---------------------------------------------------------------------------
__global__ __launch_bounds__(256)
void out_proj_kernel(const float* __restrict__ Ctx,
                     const float* __restrict__ Wo,
                     const float* __restrict__ bias,
                     float* __restrict__ Out) {
  __shared__ float Asub[64][20];
  __shared__ float Bsub[64][20];

  const int tid = threadIdx.x;
  const int wave = tid >> 5, lane = tid & 31;
  const int lhalf = lane >> 4, l16 = lane & 15;

  const int mBase = blockIdx.y * 64;
  const int nBase = blockIdx.x * 64;
  const int mOff = (wave >> 1) * 16;
  const int nOff = (wave & 1) * 32;

  v8f acc0 = {};
  v8f acc1 = {};

  const int ldRow = tid >> 2;
  const int ldCol = (tid & 3) * 4;

  for (int k0 = 0; k0 < D_MODEL; k0 += 16) {
    const float4 av =
        *(const float4*)(Ctx + (size_t)(mBase + ldRow) * D_MODEL + k0 + ldCol);
    const float4 bv =
        *(const float4*)(Wo + (size_t)(nBase + ldRow) * D_MODEL + k0 + ldCol);
    *(float4*)(&Asub[ldRow][ldCol]) = av;
    *(float4*)(&Bsub[ldRow][ldCol]) = bv;
    __syncthreads();

    if (k0 + 16 < D_MODEL) {
      __builtin_prefetch(Ctx + (size_t)(mBase + ldRow) * D_MODEL + k0 + 16 + ldCol, 0, 3);
      __builtin_prefetch(Wo + (size_t)(nBase + ldRow) * D_MODEL + k0 + 16 + ldCol, 0, 3);
    }

#pragma unroll
    for (int kk = 0; kk < 16; kk += 4) {
      const int kf = kk + 2 * lhalf;
      v2f a  = *(const v2f*)(&Asub[mOff + l16][kf]);
      v2f b0 = *(const v2f*)(&Bsub[nOff + l16][kf]);
      v2f b1 = *(const v2f*)(&Bsub[nOff + 16 + l16][kf]);
      acc0 = wmma4(a, b0, acc0);
      acc1 = wmma4(a, b1, acc1);
    }
    __syncthreads();
  }

#pragma unroll
  for (int v = 0; v < 8; ++v) {
    const int m = mBase + mOff + v + 8 * lhalf;
    {
      const int col = nBase + nOff + l16;
      Out[(size_t)m * D_MODEL + col] = acc0[v] + bias[col];
    }
    {
      const int col = nBase + nOff + 16 + l16;
      Out[(size_t)m * D_MODEL + col] = acc1[v] + bias[col];
    }
  }
}

// ---------------------------------------------------------------------------
extern "C" void kernel_launch(void* const* d_in, const int* in_sizes, int n_in,
                              void* d_out, int out_size, void* d_ws,
                              size_t ws_size, hipStream_t stream) {
  const float* X     = (const float*)d_in[0];
  const float* ts    = (const float*)d_in[1];
  const float* decay = (const float*)d_in[2];
  const unsigned char* pmask = (const unsigned char*)d_in[3];
  const float* Wq = (const float*)d_in[4];
  const float* Wk = (const float*)d_in[5];
  const float* Wv = (const float*)d_in[6];
  const float* Wo = (const float*)d_in[7];
  const float* bo = (const float*)d_in[8];
  const float* lt = (const float*)d_in[9];
  float* out = (float*)d_out;

  // workspace: Q | K | V | ctx, each B*L*D floats (16 MB) -> 64 MB total
  float* ws = (float*)d_ws;
  const size_t elems = (size_t)BATCH * SEQ * D_MODEL;
  float* Qb  = ws;
  float* Kb  = ws + elems;
  float* Vb  = ws + 2 * elems;
  float* Ctx = ws + 3 * elems;

  dim3 g1(D_MODEL / 64, (BATCH * SEQ) / 64, 3);
  qkv_proj_kernel<<<g1, 256, 0, stream>>>(X, Wq, Wk, Wv, Qb, Kb, Vb);

  dim3 g2(SEQ / 64, NHEAD, BATCH);
  attn_kernel<<<g2, 128, 0, stream>>>(Qb, Kb, Vb, ts, decay, pmask, lt, Ctx);

  dim3 g3(D_MODEL / 64, (BATCH * SEQ) / 64);
  out_proj_kernel<<<g3, 256, 0, stream>>>(Ctx, Wo, bo, out);
}